// CPC_14010183319717
// MI455X (gfx1250) — compile-verified
//
#include <hip/hip_runtime.h>

// ---------------------------------------------------------------------------
// CDNA5 (gfx1250) fp32 WMMA pipeline for the CPC reference.
//   V_WMMA_F32_16X16X4_F32:  D(16x16,f32) = A(16x4,f32) * B(4x16,f32) + C
//   Per-lane (wave32): half=lane>>4, j=lane&15
//     A: v2f = A[m=j][k0+2*half], A[m=j][k0+2*half+1]
//     B: v2f = B[k0+2*half][n=j], B[k0+2*half+1][n=j]
//     D: v8f, element i -> (m = i + 8*half, n = j)
// ---------------------------------------------------------------------------

typedef __attribute__((ext_vector_type(2))) float v2f;
typedef __attribute__((ext_vector_type(8))) float v8f;

__device__ __forceinline__ v8f wmma4(v2f a, v2f b, v8f c) {
  return __builtin_amdgcn_wmma_f32_16x16x4_f32(false, a, false, b, (short)0, c,
                                               false, false);
}

#define T_IN 128
#define T_OUT 12
#define LATENT 256
#define S_TOT 140  // T_IN + T_OUT
#define BATCH 256

// ---------------------------------------------------------------------------
// Generic WMMA GEMM: C[z] = A[z](MxK,row) * B[z](KxN,row) (+ bias)
// block = 256 threads (8 waves). Block tile: M=256 (32 rows/wave, two 16-row
// subtiles sharing each B fragment), N=64. Requires M % 256 == 0,
// N % 64 == 0, K % 4 == 0, K >= 8.
// Software-pipelined: next k-chunk's A/B fragments are loaded into registers
// before the current chunk's 8 WMMAs issue -> loads overlap matrix ops,
// and all addressing is pointer-increment (no per-iter 64-bit multiplies).
// ---------------------------------------------------------------------------
__global__ __launch_bounds__(256) void gemm_wmma_f32(
    const float* __restrict__ A, int lda, long long batchA,
    const float* __restrict__ B, int ldb, long long batchB,
    const float* __restrict__ bias, int bias_per_row, long long batchBias,
    float* __restrict__ C, int ldc, long long batchC, int M, int N, int K) {
  const int z = blockIdx.z;
  A += (long long)z * batchA;
  B += (long long)z * batchB;
  C += (long long)z * batchC;
  if (bias) bias += (long long)z * batchBias;
  (void)M;
  (void)N;

  const int lane = threadIdx.x & 31;
  const int wave = threadIdx.x >> 5;
  const int half = lane >> 4;
  const int j = lane & 15;
  const int m0 = blockIdx.y * 256 + wave * 32;
  const int n0 = blockIdx.x * 64;

  v8f acc[2][4];
#pragma unroll
  for (int s = 0; s < 2; ++s)
#pragma unroll
    for (int nt = 0; nt < 4; ++nt) acc[s][nt] = v8f{};

  const float* Ap0 = A + (long long)(m0 + j) * lda + 2 * half;
  const float* Ap1 = Ap0 + 16LL * lda;
  const float* Bp = B + (long long)(2 * half) * ldb + n0 + j;
  const long long bstep = 4LL * (long long)ldb;

  // prologue loads (k-chunk 0)
  v2f a0 = *(const v2f*)Ap0;
  v2f a1 = *(const v2f*)Ap1;
  v2f bv[4];
#pragma unroll
  for (int nt = 0; nt < 4; ++nt) {
    bv[nt].x = Bp[nt * 16];
    bv[nt].y = Bp[ldb + nt * 16];
  }

  for (int k0 = 0; k0 < K - 4; k0 += 4) {
    Ap0 += 4;
    Ap1 += 4;
    Bp += bstep;
    // prefetch next chunk while current WMMAs execute
    v2f a0n = *(const v2f*)Ap0;
    v2f a1n = *(const v2f*)Ap1;
    v2f bvn[4];
#pragma unroll
    for (int nt = 0; nt < 4; ++nt) {
      bvn[nt].x = Bp[nt * 16];
      bvn[nt].y = Bp[ldb + nt * 16];
    }
#pragma unroll
    for (int nt = 0; nt < 4; ++nt) {
      acc[0][nt] = wmma4(a0, bv[nt], acc[0][nt]);
      acc[1][nt] = wmma4(a1, bv[nt], acc[1][nt]);
    }
    a0 = a0n;
    a1 = a1n;
#pragma unroll
    for (int nt = 0; nt < 4; ++nt) bv[nt] = bvn[nt];
  }
#pragma unroll
  for (int nt = 0; nt < 4; ++nt) {
    acc[0][nt] = wmma4(a0, bv[nt], acc[0][nt]);
    acc[1][nt] = wmma4(a1, bv[nt], acc[1][nt]);
  }

#pragma unroll
  for (int sub = 0; sub < 2; ++sub) {
#pragma unroll
    for (int nt = 0; nt < 4; ++nt) {
      const int n = n0 + nt * 16 + j;
      const float bcol = bias ? (bias_per_row ? 0.0f : bias[n]) : 0.0f;
      float* Cp = C + (long long)(m0 + sub * 16 + 8 * half) * ldc + n;
#pragma unroll
      for (int i = 0; i < 8; ++i) {
        float v = acc[sub][nt][i] + bcol;
        if (bias && bias_per_row) v += bias[m0 + sub * 16 + 8 * half + i];
        Cp[(long long)i * ldc] = v;
      }
    }
  }
}

// ---------------------------------------------------------------------------
// Conv1d (kernel=stride=4, VALID) as WMMA GEMM. X is (B,L,C) row-major, so the
// 48-element window for output s is contiguous at X + b*24576 + 48*s.
// Wt is the permuted weight (48 x 256): Wt[t*12+c][m] = conv_w[m][c][t].
// One wave per block: 16 s-rows x 64 latent cols; grid (4, 9, 256).
// enc layout: (s, b, m) -> enc[s*65536 + b*256 + m], only s < 140 stored.
// ---------------------------------------------------------------------------
__global__ __launch_bounds__(32) void conv_wmma(const float* __restrict__ X,
                                                const float* __restrict__ Wt,
                                                const float* __restrict__ bias,
                                                float* __restrict__ enc) {
  const int lane = threadIdx.x & 31;
  const int half = lane >> 4;
  const int j = lane & 15;
  const int m0 = blockIdx.x * 64;
  const int s0 = blockIdx.y * 16;
  const int b = blockIdx.z;

  int s = s0 + j;
  if (s > S_TOT - 1) s = S_TOT - 1;  // clamp loads; stores masked
  const float* Ap = X + (long long)b * 24576 + (long long)s * 48 + 2 * half;
  const float* Bp = Wt + (2 * half) * 256 + m0 + j;

  v8f acc[4] = {v8f{}, v8f{}, v8f{}, v8f{}};
  for (int k0 = 0; k0 < 48; k0 += 4) {
    v2f a = *(const v2f*)Ap;
    v2f bv[4];
#pragma unroll
    for (int nt = 0; nt < 4; ++nt) {
      bv[nt].x = Bp[nt * 16];
      bv[nt].y = Bp[256 + nt * 16];
    }
#pragma unroll
    for (int nt = 0; nt < 4; ++nt) acc[nt] = wmma4(a, bv[nt], acc[nt]);
    Ap += 4;
    Bp += 1024;  // 4 rows of 256
  }

#pragma unroll
  for (int nt = 0; nt < 4; ++nt) {
    const int n = m0 + nt * 16 + j;
    const float bv = bias[n];
#pragma unroll
    for (int i = 0; i < 8; ++i) {
      const int so = s0 + i + 8 * half;
      if (so < S_TOT)
        enc[((long long)so * 256 + b) * 256 + n] = acc[nt][i] + bv;
    }
  }
}

// ----------------------------- small helpers -------------------------------
__global__ void zero_f32(float* p, int n) {
  int i = blockIdx.x * blockDim.x + threadIdx.x;
  if (i < n) p[i] = 0.0f;
}

// out[c*R + r] = in[r*C + c]   (in is R x C)
__global__ void transpose_f32(const float* __restrict__ in,
                              float* __restrict__ out, int R, int C) {
  int i = blockIdx.x * blockDim.x + threadIdx.x;
  if (i < R * C) {
    int r = i / C, c = i % C;
    out[(long long)c * R + r] = in[i];
  }
}

// Wt[(t*12+c)*256 + m] = conv_w[m*48 + c*4 + t]
__global__ void permute_convw(const float* __restrict__ w,
                              float* __restrict__ out) {
  int i = blockIdx.x * blockDim.x + threadIdx.x;
  if (i < 256 * 48) {
    int m = i / 48, rem = i % 48;
    int c = rem / 4, t = rem % 4;
    out[(t * 12 + c) * 256 + m] = w[i];
  }
}

// GRU gating: h' = (1-z)*n + z*h, gates from gi/gh (256 x 768 each)
__global__ __launch_bounds__(256) void gru_gate(const float* __restrict__ gi,
                                                const float* __restrict__ gh,
                                                const float* __restrict__ hin,
                                                float* __restrict__ hout) {
  int idx = blockIdx.x * blockDim.x + threadIdx.x;
  if (idx >= 256 * 256) return;
  int b = idx >> 8, m = idx & 255;
  const float* gib = gi + b * 768;
  const float* ghb = gh + b * 768;
  float r = 1.0f / (1.0f + __expf(-(gib[m] + ghb[m])));
  float z = 1.0f / (1.0f + __expf(-(gib[256 + m] + ghb[256 + m])));
  float n = tanhf(gib[512 + m] + r * ghb[512 + m]);
  hout[idx] = (1.0f - z) * n + z * hin[idx];
}

// Pass 1: online (max, argmax, sum-exp) over 256-row chunks, per column.
// Thread t owns column t -> coalesced row-major reads. grid = 140 chunks.
__global__ __launch_bounds__(256) void col_reduce_pass1(
    const float* __restrict__ sim, float* __restrict__ pmax,
    int* __restrict__ parg, float* __restrict__ psum) {
  const int c = threadIdx.x;
  const int rbase = blockIdx.x * 256;
  float m = -__builtin_inff(), s = 0.0f;
  int arg = rbase;
  for (int i = 0; i < 256; ++i) {
    const int r = rbase + i;
    float x = sim[(long long)r * 256 + c];
    if (x > m) {
      s = s * __expf(m - x) + 1.0f;
      m = x;
      arg = r;
    } else {
      s += __expf(x - m);
    }
  }
  const int o = rbase + c;
  pmax[o] = m;
  parg[o] = arg;
  psum[o] = s;
}

// Pass 2: merge 140 chunk partials per column (first-occurrence argmax),
// then emit per-(k,c) loss part (diag - lse) and argmax-hit flag.
__global__ __launch_bounds__(256) void col_reduce_pass2(
    const float* __restrict__ sim, const float* __restrict__ pmax,
    const int* __restrict__ parg, const float* __restrict__ psum,
    float* __restrict__ lossp, float* __restrict__ hitp, int k) {
  const int c = threadIdx.x;
  float m = -__builtin_inff(), s = 0.0f;
  int arg = 0;
  for (int ch = 0; ch < S_TOT; ++ch) {
    const int o = ch * 256 + c;
    float m2 = pmax[o], s2 = psum[o];
    int a2 = parg[o];
    if (m2 > m) {
      s = s * __expf(m - m2) + s2;
      m = m2;
      arg = a2;
    } else {
      s += s2 * __expf(m2 - m);
    }
  }
  const float lse = m + logf(s);
  const int drow = (T_IN + k) * 256 + c;
  const float dval = sim[(long long)drow * 256 + c];
  lossp[k * 256 + c] = dval - lse;
  hitp[k * 256 + c] = (arg == drow) ? 1.0f : 0.0f;
}

__global__ __launch_bounds__(256) void finalize(const float* __restrict__ lossp,
                                                const float* __restrict__ hitp,
                                                float* __restrict__ out) {
  __shared__ float sl[256], sa[256];
  const int t = threadIdx.x;
  float l = 0.0f, a = 0.0f;
  for (int k = 0; k < T_OUT; ++k) {
    l += lossp[k * 256 + t];
    a += hitp[k * 256 + t];
  }
  sl[t] = l;
  sa[t] = a;
  __syncthreads();
  for (int off = 128; off > 0; off >>= 1) {
    if (t < off) {
      sl[t] += sl[t + off];
      sa[t] += sa[t + off];
    }
    __syncthreads();
  }
  if (t == 0) {
    out[0] = sa[0] / 3072.0f;   // accuracy
    out[1] = -sl[0] / 3072.0f;  // loss
  }
}

__global__ void copy_hidden(const float* __restrict__ h,
                            float* __restrict__ out) {
  int i = blockIdx.x * blockDim.x + threadIdx.x;
  if (i < 256 * 256) out[2 + i] = h[i];
}

// ---------------------------------------------------------------------------
extern "C" void kernel_launch(void* const* d_in, const int* in_sizes, int n_in,
                              void* d_out, int out_size, void* d_ws,
                              size_t ws_size, hipStream_t stream) {
  (void)in_sizes; (void)n_in; (void)out_size;
  const float* X      = (const float*)d_in[0];  // (256,2048,12)
  const float* conv_w = (const float*)d_in[1];  // (256,12,4)
  const float* conv_b = (const float*)d_in[2];  // (256)
  const float* W_ih   = (const float*)d_in[3];  // (768,256)
  const float* W_hh   = (const float*)d_in[4];  // (768,256)
  const float* b_ih   = (const float*)d_in[5];  // (768)
  const float* b_hh   = (const float*)d_in[6];  // (768)
  const float* pred_W = (const float*)d_in[7];  // (12,256,256)
  const float* pred_b = (const float*)d_in[8];  // (12,256)
  float* out = (float*)d_out;  // [accuracy, loss, hidden(65536)]

  // ---- workspace carve-up (~81 MB base; +101 MB optional gi_all) ----
  float* ws = (float*)d_ws;
  size_t off = 0;
  auto alloc = [&](size_t n) {
    float* p = ws + off;
    off += (n + 3) & ~(size_t)3;
    return p;
  };
  float* enc   = alloc((size_t)S_TOT * 256 * 256);  // (s,b,m)
  float* cWT   = alloc(48 * 256);
  float* WTih  = alloc(256 * 768);
  float* WThh  = alloc(256 * 768);
  float* hA    = alloc(256 * 256);
  float* hB    = alloc(256 * 256);
  float* Gi    = alloc(256 * 768);
  float* Gh    = alloc(256 * 768);
  float* ccT   = alloc(256 * 256);
  float* predB = alloc((size_t)T_OUT * 256 * 256);  // (k,m,c)
  float* simb  = alloc((size_t)S_TOT * 256 * 256);  // (s*256+b, c), reused /k
  float* pmax  = alloc(S_TOT * 256);
  float* psum  = alloc(S_TOT * 256);
  int*  parg   = (int*)alloc(S_TOT * 256);
  float* lossp = alloc(T_OUT * 256);
  float* hitp  = alloc(T_OUT * 256);
  // optional: gi for all 128 steps in one parallel GEMM (no recurrence on x)
  const size_t giAllN = (size_t)T_IN * 256 * 768;
  const bool useGiAll = (off + giAllN) * sizeof(float) <= ws_size;
  float* giAll = useGiAll ? alloc(giAllN) : nullptr;

  // ---- weight transforms (row-major coalesced B operands for every GEMM) --
  permute_convw<<<48, 256, 0, stream>>>(conv_w, cWT);
  transpose_f32<<<(768 * 256 + 255) / 256, 256, 0, stream>>>(W_ih, WTih, 768, 256);
  transpose_f32<<<(768 * 256 + 255) / 256, 256, 0, stream>>>(W_hh, WThh, 768, 256);

  // ---- conv encoder (only the 140 consumed timesteps) ----
  conv_wmma<<<dim3(4, 9, 256), 32, 0, stream>>>(X, cWT, conv_b, enc);

  // ---- GRU: gi has no recurrence -> one big GEMM when ws allows; only the
  //      gh GEMM (256x768x256) + gating stay on the 128-step serial chain ----
  zero_f32<<<256, 256, 0, stream>>>(hA, 256 * 256);
  if (useGiAll) {
    // (128*256) x 768 x 256; enc rows (t,b) are exactly its first 32768 rows
    gemm_wmma_f32<<<dim3(12, T_IN, 1), 256, 0, stream>>>(
        enc, 256, 0, WTih, 768, 0, b_ih, 0, 0, giAll, 768, 0, T_IN * 256, 768,
        256);
  }
  float* hin = hA;
  float* hout = hB;
  for (int t = 0; t < T_IN; ++t) {
    float* gi_t;
    if (useGiAll) {
      gi_t = giAll + (size_t)t * 256 * 768;
    } else {
      gi_t = Gi;
      gemm_wmma_f32<<<dim3(12, 1, 1), 256, 0, stream>>>(
          enc + (size_t)t * 65536, 256, 0, WTih, 768, 0, b_ih, 0, 0, Gi, 768,
          0, 256, 768, 256);
    }
    gemm_wmma_f32<<<dim3(12, 1, 1), 256, 0, stream>>>(
        hin, 256, 0, WThh, 768, 0, b_hh, 0, 0, Gh, 768, 0, 256, 768, 256);
    gru_gate<<<256, 256, 0, stream>>>(gi_t, Gh, hin, hout);
    float* tmp = hin; hin = hout; hout = tmp;
  }
  float* hfin = hin;  // h_last == context[-1]

  // ---- predB[k](m,c) = pred_W[k](m,l) @ ccT(l,c) + pred_b[k][m] ----
  transpose_f32<<<256, 256, 0, stream>>>(hfin, ccT, 256, 256);
  gemm_wmma_f32<<<dim3(4, 1, T_OUT), 256, 0, stream>>>(
      pred_W, 256, 65536, ccT, 256, 0, pred_b, 1, 256, predB, 256, 65536, 256,
      256, 256);

  // ---- per-k: sim GEMM (35840 x 256 x 256) + column log-softmax/argmax ----
  for (int k = 0; k < T_OUT; ++k) {
    gemm_wmma_f32<<<dim3(4, 140, 1), 256, 0, stream>>>(
        enc, 256, 0, predB + (size_t)k * 65536, 256, 0, nullptr, 0, 0, simb,
        256, 0, S_TOT * 256, 256, 256);
    col_reduce_pass1<<<S_TOT, 256, 0, stream>>>(simb, pmax, parg, psum);
    col_reduce_pass2<<<1, 256, 0, stream>>>(simb, pmax, parg, psum, lossp,
                                            hitp, k);
  }

  finalize<<<1, 256, 0, stream>>>(lossp, hitp, out);
  copy_hidden<<<256, 256, 0, stream>>>(hfin, out);
}